// FeatureExtractionHsi_69810398429335
// MI455X (gfx1250) — compile-verified
//
#include <hip/hip_runtime.h>
#include <math.h>

typedef __attribute__((ext_vector_type(2))) float v2f;
typedef __attribute__((ext_vector_type(8))) float v8f;

#define HSI_C   64
#define HSI_HW  11
#define HSI_SP  121          // 11*11
#define OUT_CH  100
#define GEN_OUT 6400
#define M_TILE  16
#define A_MSTR  68           // padded m-stride (floats) -> conflict-free WMMA A loads
#define EPS     1e-5f

// grid: (7 m-tiles, B samples), block: 256 threads (8 waves)
__global__ __launch_bounds__(256)
void FeatureExtractionHsi_69810398429335_kernel(
    const float* __restrict__ x_in,     // [B,64,11,11]
    const float* __restrict__ ge_w,     // [6400,2,3,3]
    const float* __restrict__ ge_b,     // [6400]
    const float* __restrict__ bn_gamma, // [6400]
    const float* __restrict__ bn_beta,  // [6400]
    const float* __restrict__ bn_mean,  // [6400]
    const float* __restrict__ bn_var,   // [6400]
    float* __restrict__ out)            // [B,100,11,11]
{
    __shared__ float xH[2 * HSI_SP];              // pooled map
    __shared__ float A_lds[9 * M_TILE * A_MSTR];  // A[tap][m][c], c-minor

    const int tid    = threadIdx.x;
    const int batch  = blockIdx.y;
    const int o_base = blockIdx.x * M_TILE;

    const float* xs = x_in + (size_t)batch * HSI_C * HSI_SP;

    // ---------- Phase 0: channel pooling (avg + max) ----------
    for (int p = tid; p < HSI_SP; p += 256) {
        float s = 0.f, mx = -3.4e38f;
        #pragma unroll 8
        for (int c = 0; c < HSI_C; ++c) {
            float v = xs[c * HSI_SP + p];
            s += v;
            mx = fmaxf(mx, v);
        }
        xH[p]          = s * (1.0f / HSI_C);
        xH[HSI_SP + p] = mx;
    }
    __syncthreads();

    // ---------- Phase 1: generate per-sample filters for this M-tile ----------
    // 16 out-channels * 64 in-channels = 1024 gen channels; 4 per thread.
    #pragma unroll
    for (int v = 0; v < 4; ++v) {
        int gl = v * 256 + tid;                 // 0..1023
        int g  = o_base * HSI_C + gl;           // global gen channel
        bool gv = (g < GEN_OUT);
        int gc = gv ? g : 0;

        float w[18];
        #pragma unroll
        for (int t = 0; t < 18; ++t) w[t] = ge_w[gc * 18 + t];
        float bias = ge_b[gc];
        float mean = bn_mean[gc];
        float bet  = bn_beta[gc];
        float invs = bn_gamma[gc] * rsqrtf(bn_var[gc] + EPS);

        int m = gl >> 6;        // local out-channel row
        int c = gl & 63;        // in-channel (K-minor)

        #pragma unroll
        for (int tap = 0; tap < 9; ++tap) {
            int oy = tap / 3, ox = tap % 3;
            float acc = bias;
            #pragma unroll
            for (int cin = 0; cin < 2; ++cin)
                #pragma unroll
                for (int th = 0; th < 3; ++th)
                    #pragma unroll
                    for (int tw = 0; tw < 3; ++tw)
                        acc += w[cin * 9 + th * 3 + tw] *
                               xH[cin * HSI_SP + (4 * oy + th) * HSI_HW + (4 * ox + tw)];
            acc = (acc - mean) * invs + bet;    // BatchNorm (eval)
            acc = fmaxf(acc, 0.f);              // ReLU
            A_lds[tap * (M_TILE * A_MSTR) + m * A_MSTR + c] = gv ? acc : 0.f;
        }
    }
    __syncthreads();

    // ---------- Phase 2: dynamic conv as implicit GEMM via f32 WMMA ----------
    // M=16 out-ch, N=121 positions (8 tiles of 16, one per wave), K=9 taps * 64 ch.
    const int lane  = tid & 31;
    const int lhalf = lane >> 4;      // 0: lanes 0-15, 1: lanes 16-31
    const int lmod  = lane & 15;
    const int ntile = tid >> 5;       // wave id = N-tile
    const int n     = ntile * 16 + lmod;
    const bool nvalid = (n < HSI_SP);
    int pi = nvalid ? (n / HSI_HW) : 0;
    int pj = nvalid ? (n % HSI_HW) : 0;

    v8f acc = {0.f, 0.f, 0.f, 0.f, 0.f, 0.f, 0.f, 0.f};

    for (int tap = 0; tap < 9; ++tap) {
        int kh = tap / 3, kw = tap % 3;
        int y  = pi + kh - 1;
        int xx = pj + kw - 1;
        // Clamped address is always in-bounds; border zeros realized via mask
        // multiply so all loads are unconditional (no exec-mask predication).
        bool bvalid = nvalid && (y >= 0) && (y < HSI_HW) && (xx >= 0) && (xx < HSI_HW);
        float bmask = bvalid ? 1.0f : 0.0f;
        int pos = bvalid ? (y * HSI_HW + xx) : 0;

        // B fragment: lane covers channels c = 4s + 2*lhalf + {0,1} at spatial 'pos'
        const float* bp = xs + pos + (2 * lhalf) * HSI_SP;
        // A fragment: A[tap][m=lmod][c = 4s + 2*lhalf + {0,1}]
        const float* ap = &A_lds[tap * (M_TILE * A_MSTR) + lmod * A_MSTR + 2 * lhalf];

        #pragma unroll
        for (int s = 0; s < 16; ++s) {
            v2f b;
            b.x = bp[s * 4 * HSI_SP]          * bmask;
            b.y = bp[s * 4 * HSI_SP + HSI_SP] * bmask;
            v2f a = *(const v2f*)(ap + 4 * s);
            acc = __builtin_amdgcn_wmma_f32_16x16x4_f32(
                      false, a, false, b, (short)0, acc, false, false);
        }
    }

    // ---------- Store: C/D layout lane-lo rows 0..7, lane-hi rows 8..15 ----------
    float* yout = out + (size_t)batch * OUT_CH * HSI_SP;
    if (nvalid) {
        #pragma unroll
        for (int r = 0; r < 8; ++r) {
            int m = r + 8 * lhalf;
            int o = o_base + m;
            if (o < OUT_CH) yout[o * HSI_SP + n] = acc[r];
        }
    }
}

extern "C" void kernel_launch(void* const* d_in, const int* in_sizes, int n_in,
                              void* d_out, int out_size, void* d_ws, size_t ws_size,
                              hipStream_t stream) {
    const float* x_in     = (const float*)d_in[0];
    const float* ge_w_w   = (const float*)d_in[1];
    const float* ge_w_b   = (const float*)d_in[2];
    const float* bn_gamma = (const float*)d_in[3];
    const float* bn_beta  = (const float*)d_in[4];
    const float* bn_mean  = (const float*)d_in[5];
    const float* bn_var   = (const float*)d_in[6];
    float* out = (float*)d_out;

    int batch = in_sizes[0] / (HSI_C * HSI_SP);   // 1024
    dim3 grid((OUT_CH + M_TILE - 1) / M_TILE, batch);  // (7, B)
    FeatureExtractionHsi_69810398429335_kernel<<<grid, 256, 0, stream>>>(
        x_in, ge_w_w, ge_w_b, bn_gamma, bn_beta, bn_mean, bn_var, out);
}